// HausdorffLoss_42021960024228
// MI455X (gfx1250) — compile-verified
//
#include <hip/hip_runtime.h>

typedef __attribute__((ext_vector_type(2))) float v2f;
typedef __attribute__((ext_vector_type(8))) float v8f;

#define WPB  4    // waves per block
#define TILE 16
#define MT   2    // 16-row query tiles per wave (reuse each streamed B tile)

// raw v_min_num_f32: skip fminf's NaN-canonicalize ops (inputs are finite)
__device__ __forceinline__ float fmin_raw(float a, float b) {
    float r;
    asm("v_min_num_f32 %0, %1, %2" : "=v"(r) : "v"(a), "v"(b));
    return r;
}

// DPP16 row rotate (within each 16-lane half of the wave)
template <int CTRL>
__device__ __forceinline__ float row_rot(float x) {
    int v = __builtin_amdgcn_update_dpp(0, __builtin_bit_cast(int, x),
                                        CTRL, 0xF, 0xF, true);
    return __builtin_bit_cast(float, v);
}

// min across the 16 lanes of each half: row_ror 8/4/2/1 (pure VALU, no LDS)
__device__ __forceinline__ float row16_min(float x) {
    x = fmin_raw(x, row_rot<0x128>(x));   // row_ror:8
    x = fmin_raw(x, row_rot<0x124>(x));   // row_ror:4
    x = fmin_raw(x, row_rot<0x122>(x));   // row_ror:2
    x = fmin_raw(x, row_rot<0x121>(x));   // row_ror:1
    return x;
}

__global__ void hd_zero_kernel(float* out, int n) {
    int i = blockIdx.x * blockDim.x + threadIdx.x;
    if (i < n) out[i] = 0.0f;
}

// Directed Chamfer term via V_WMMA_F32_16X16X4_F32.
// A[m] = (q0,q1,q2,1), B[n] = (-2t0,-2t1,-2t2,||t||^2), C[m][n] = ||q_m||^2
// => D[m][n] = ||q_m - t_n||^2 computed entirely by the matrix pipe.
__global__ __launch_bounds__(WPB * 32)
void hd_chamfer_dir_kernel(const float* __restrict__ qry,
                           const float* __restrict__ tgt,
                           float* __restrict__ out, int M, int N) {
    const int lane  = threadIdx.x & 31;
    const int wave  = threadIdx.x >> 5;
    const int b     = blockIdx.y;
    const int m0    = (blockIdx.x * WPB + wave) * (TILE * MT);

    const float* q = qry + (size_t)b * (size_t)M * 3u;
    const float* t = tgt + (size_t)b * (size_t)N * 3u;

    const int p     = lane & 15;      // position within 16-lane half
    const int khalf = lane >> 4;      // 0: K={0,1}; 1: K={2,3}

    // ---- prologue: A fragments + loop-invariant C fragments (= ||q||^2) ----
    v2f afrag[MT];
    v8f cfrag[MT];
    #pragma unroll
    for (int tq = 0; tq < MT; ++tq) {
        int row  = m0 + tq * TILE + p;
        float q0 = q[row * 3 + 0];
        float q1 = q[row * 3 + 1];
        float q2 = q[row * 3 + 2];
        afrag[tq].x = khalf ? q2 : q0;
        afrag[tq].y = khalf ? 1.0f : q1;            // K=3 slot multiplies ||t||^2
        float sq1 = q0 * q0 + q1 * q1 + q2 * q2;    // lane holds tile-row p
        // C/D layout: reg r on half h is row (r + 8h); broadcast from lane r+8h
        #pragma unroll
        for (int r = 0; r < 8; ++r)
            cfrag[tq][r] = __shfl(sq1, r + 8 * khalf, 32);
    }

    v8f tmin[MT];
    #pragma unroll
    for (int tq = 0; tq < MT; ++tq)
        #pragma unroll
        for (int r = 0; r < 8; ++r) tmin[tq][r] = 3.0e38f;

    // ---- software-pipelined main loop: 16 target points per iteration ----
    // Prefetch tile 0, then inside the loop issue the load for tile i+1
    // before consuming tile i, so s_wait_loadcnt lands after the WMMA+min
    // chain instead of right after the load (only ~1 wave/SIMD here, so
    // in-wave ILP is the only latency cover).
    float ct0 = t[p * 3 + 0];
    float ct1 = t[p * 3 + 1];
    float ct2 = t[p * 3 + 2];

    #pragma unroll 2
    for (int n0 = 0; n0 < N; n0 += TILE) {
        int nn   = n0 + TILE;
        int ncol = (nn < N ? nn : 0) + p;   // wrap on final iter (value unused)
        float pt0 = t[ncol * 3 + 0];
        float pt1 = t[ncol * 3 + 1];
        float pt2 = t[ncol * 3 + 2];

        float sq2 = ct0 * ct0 + ct1 * ct1 + ct2 * ct2;
        v2f bfrag;
        bfrag.x = -2.0f * (khalf ? ct2 : ct0);      // K=0 / K=2 row of B
        bfrag.y = khalf ? sq2 : (-2.0f * ct1);      // K=3 carries ||t||^2

        #pragma unroll
        for (int tq = 0; tq < MT; ++tq) {
            v8f d2 = __builtin_amdgcn_wmma_f32_16x16x4_f32(
                false, afrag[tq], false, bfrag, (short)0, cfrag[tq],
                false, false);
            #pragma unroll
            for (int r = 0; r < 8; ++r)
                tmin[tq][r] = fmin_raw(tmin[tq][r], d2[r]);
        }

        ct0 = pt0; ct1 = pt1; ct2 = pt2;
    }

    // ---- epilogue: min across 16 lanes per half (DPP), sqrt, row sum ----
    float acc = 0.0f;
    #pragma unroll
    for (int tq = 0; tq < MT; ++tq) {
        #pragma unroll
        for (int r = 0; r < 8; ++r) {
            float mn = row16_min(tmin[tq][r]);       // min_n d^2 for row r+8h
            mn = mn > 0.0f ? mn : 0.0f;
            acc += __builtin_amdgcn_sqrtf(mn);
        }
    }
    acc += __shfl_xor(acc, 16, 32);                  // halves hold rows 0-7 / 8-15

    if (lane == 0) atomicAdd(&out[b], acc * (1.0f / (float)M));
}

extern "C" void kernel_launch(void* const* d_in, const int* in_sizes, int n_in,
                              void* d_out, int out_size, void* d_ws, size_t ws_size,
                              hipStream_t stream) {
    const float* set1 = (const float*)d_in[0];
    const float* set2 = (const float*)d_in[1];
    float* out = (float*)d_out;

    const int B = out_size;                    // (B,) float32 output
    const int M = in_sizes[0] / (3 * B);       // 4096
    const int N = in_sizes[1] / (3 * B);       // 4096

    hd_zero_kernel<<<1, 64, 0, stream>>>(out, out_size);

    dim3 blk(WPB * 32);
    const int rows_per_block = TILE * MT * WPB;   // 128

    dim3 g1(M / rows_per_block, B);            // set1 -> set2 direction
    hd_chamfer_dir_kernel<<<g1, blk, 0, stream>>>(set1, set2, out, M, N);

    dim3 g2(N / rows_per_block, B);            // set2 -> set1 direction
    hd_chamfer_dir_kernel<<<g2, blk, 0, stream>>>(set2, set1, out, N, M);
}